// GCN_1580547975274
// MI455X (gfx1250) — compile-verified
//
#include <hip/hip_runtime.h>
#include <hip/hip_bf16.h>

#define FIN  4
#define FHID 64

typedef float v2f __attribute__((ext_vector_type(2)));
typedef float v8f __attribute__((ext_vector_type(8)));

static __host__ int cdiv(int a, int b) { return (a + b - 1) / b; }

// ---------------------------------------------------------------------------
// Degree: deg[dst] += 1 for every edge incl. self loops (e in [E, E+N))
// ---------------------------------------------------------------------------
__global__ void k_degree(const int* __restrict__ ei, float* __restrict__ deg,
                         int E, int N) {
    int e = blockIdx.x * blockDim.x + threadIdx.x;
    if (e >= E + N) return;
    int dst = (e < E) ? ei[E + e] : (e - E);
    atomicAdd(&deg[dst], 1.0f);
}

// dis = deg > 0 ? deg^-0.5 : 0   (in place)
__global__ void k_rsqrt(float* __restrict__ d, int N) {
    int n = blockIdx.x * blockDim.x + threadIdx.x;
    if (n >= N) return;
    float v = d[n];
    d[n] = (v > 0.0f) ? rsqrtf(v) : 0.0f;
}

// ---------------------------------------------------------------------------
// t = x @ W1  via V_WMMA_F32_16X16X4_F32.
// One wave computes a 16(row) x 64(col) tile: 4 WMMAs, K = FIN = 4 exactly.
// A layout (ISA 7.12.2, 32-bit A 16x4): lane l<16 -> M=l, K={0,1};
//                                       lane l>=16 -> M=l-16, K={2,3}.
// D layout: VGPR i -> M = 8*half + i, N = lane&15.
// ---------------------------------------------------------------------------
__global__ void k_xw1_wmma(const float* __restrict__ x,
                           const float* __restrict__ W1,
                           float* __restrict__ t, int N) {
    const int lane = threadIdx.x & 31;
    const int wave = threadIdx.x >> 5;
    const int tile = blockIdx.x * (blockDim.x >> 5) + wave;   // 16-row tile id
    if (tile * 16 >= N) return;

    const int half = lane >> 4;       // 0: lanes 0-15, 1: lanes 16-31
    const int l16  = lane & 15;
    const int row  = tile * 16 + l16; // A-matrix M index
    const int kb   = half * 2;        // K base held by this lane half

    v2f a;
    a.x = x[row * FIN + kb + 0];
    a.y = x[row * FIN + kb + 1];

#pragma unroll
    for (int c = 0; c < FHID / 16; ++c) {     // four 16-wide column tiles
        const int col = c * 16 + l16;         // B-matrix N index
        v2f b;
        b.x = W1[(kb + 0) * FHID + col];
        b.y = W1[(kb + 1) * FHID + col];
        v8f acc = {};
        acc = __builtin_amdgcn_wmma_f32_16x16x4_f32(
            /*neg_a=*/false, a, /*neg_b=*/false, b,
            /*c_mod=*/(short)0, acc, /*reuse_a=*/false, /*reuse_b=*/false);
#pragma unroll
        for (int i = 0; i < 8; ++i) {
            int m = half * 8 + i;
            t[(size_t)(tile * 16 + m) * FHID + col] = acc[i];
        }
    }
}

// ---------------------------------------------------------------------------
// Layer-1 edge pass: agg[dst] += t[src] * dis[src]*dis[dst]
// One wave per edge; lane handles 2 features (float2 gather, 2 f32 atomics).
// ---------------------------------------------------------------------------
__global__ void k_edge1(const int* __restrict__ ei,
                        const float* __restrict__ t,
                        const float* __restrict__ dis,
                        float* __restrict__ agg, int E, int N) {
    const int  lane = threadIdx.x & 31;
    const long long wid =
        (long long)blockIdx.x * (blockDim.x >> 5) + (threadIdx.x >> 5);
    if (wid >= (long long)E + N) return;

    int src, dst;
    if (wid < E) { src = ei[wid]; dst = ei[E + wid]; }
    else         { src = dst = (int)(wid - E); }

    const float nrm = dis[src] * dis[dst];
    const float2 v  = *(const float2*)(t + (size_t)src * FHID + lane * 2);
    float* ap = agg + (size_t)dst * FHID + lane * 2;
    atomicAdd(ap + 0, v.x * nrm);
    atomicAdd(ap + 1, v.y * nrm);
}

// ---------------------------------------------------------------------------
// z[n] = relu(agg[n] + b1) . W2   (wave per node, shfl_xor tree reduction)
// ---------------------------------------------------------------------------
__global__ void k_relu_mv(const float* __restrict__ agg,
                          const float* __restrict__ b1,
                          const float* __restrict__ W2,
                          float* __restrict__ z, int N) {
    const int lane = threadIdx.x & 31;
    const int n = blockIdx.x * (blockDim.x >> 5) + (threadIdx.x >> 5);
    if (n >= N) return;
    float s = 0.0f;
#pragma unroll
    for (int j = 0; j < FHID / 32; ++j) {
        int f = j * 32 + lane;
        float h = agg[(size_t)n * FHID + f] + b1[f];
        h = fmaxf(h, 0.0f);
        s = fmaf(h, W2[f], s);
    }
#pragma unroll
    for (int off = 16; off > 0; off >>= 1)
        s += __shfl_xor(s, off, 32);
    if (lane == 0) z[n] = s;
}

// ---------------------------------------------------------------------------
// Layer-2 edge pass: scalar messages. One thread per edge.
// ---------------------------------------------------------------------------
__global__ void k_edge2(const int* __restrict__ ei,
                        const float* __restrict__ z,
                        const float* __restrict__ dis,
                        float* __restrict__ oacc, int E, int N) {
    long long e = (long long)blockIdx.x * blockDim.x + threadIdx.x;
    if (e >= (long long)E + N) return;
    int src, dst;
    if (e < E) { src = ei[e]; dst = ei[E + e]; }
    else       { src = dst = (int)(e - E); }
    atomicAdd(&oacc[dst], z[src] * dis[src] * dis[dst]);
}

__global__ void k_sigmoid(const float* __restrict__ oacc,
                          const float* __restrict__ b2,
                          float* __restrict__ out, int N) {
    int n = blockIdx.x * blockDim.x + threadIdx.x;
    if (n >= N) return;
    float v = oacc[n] + b2[0];
    out[n] = 1.0f / (1.0f + __expf(-v));
}

// ---------------------------------------------------------------------------
extern "C" void kernel_launch(void* const* d_in, const int* in_sizes, int n_in,
                              void* d_out, int out_size, void* d_ws, size_t ws_size,
                              hipStream_t stream) {
    const float* x  = (const float*)d_in[0];
    const int*   ei = (const int*)d_in[1];   // [2, E] row-major: src then dst
    const float* W1 = (const float*)d_in[2]; // [FIN, FHID]
    const float* b1 = (const float*)d_in[3];
    const float* W2 = (const float*)d_in[4]; // [FHID, 1]
    const float* b2 = (const float*)d_in[5];
    float* out = (float*)d_out;

    const int N = in_sizes[0] / FIN;
    const int E = in_sizes[1] / 2;
    const int T = E + N;                     // edges incl. self loops

    // scratch layout (slices back to back, no alignment padding)
    char* ws = (char*)d_ws;
    float* dis  = (float*)ws;                 ws += (size_t)N * sizeof(float);
    float* t    = (float*)ws;                 ws += (size_t)N * FHID * sizeof(float);
    float* agg  = (float*)ws;                 ws += (size_t)N * FHID * sizeof(float);
    float* z    = (float*)ws;                 ws += (size_t)N * sizeof(float);
    float* oacc = (float*)ws;

    // zero accumulators every call (deterministic across graph replays)
    hipMemsetAsync(dis,  0, (size_t)N * sizeof(float), stream);
    hipMemsetAsync(agg,  0, (size_t)N * FHID * sizeof(float), stream);
    hipMemsetAsync(oacc, 0, (size_t)N * sizeof(float), stream);

    const int BT = 256;          // 8 waves / block
    const int WPB = BT / 32;

    k_degree  <<<cdiv(T, BT), BT, 0, stream>>>(ei, dis, E, N);
    k_rsqrt   <<<cdiv(N, BT), BT, 0, stream>>>(dis, N);

    const int tiles = cdiv(N, 16);
    k_xw1_wmma<<<cdiv(tiles, WPB), BT, 0, stream>>>(x, W1, t, N);

    k_edge1   <<<cdiv(T, WPB), BT, 0, stream>>>(ei, t, dis, agg, E, N);
    k_relu_mv <<<cdiv(N, WPB), BT, 0, stream>>>(agg, b1, W2, z, N);
    k_edge2   <<<cdiv(T, BT), BT, 0, stream>>>(ei, z, dis, oacc, E, N);
    k_sigmoid <<<cdiv(N, BT), BT, 0, stream>>>(oacc, b2, out, N);
}